// HashEncoder_85658827751642
// MI455X (gfx1250) — compile-verified
//
#include <hip/hip_runtime.h>
#include <cstdint>
#include <cstddef>

typedef float v2f __attribute__((ext_vector_type(2)));
typedef float v4f __attribute__((ext_vector_type(4)));

#define NLVL 16
#define LOG2T 19
#define TMASK ((1u << LOG2T) - 1u)

// res_i = float32(floor(16 * (2^(1/3))^i)) -- static per the reference's module
// constants (BASE_RES=16, FINEST_RES=512, L=16). Verified safe at the exact
// power-of-two levels (fp32(2^(1/3)) > exact, so powers land just above 2^k).
__device__ constexpr float kRes[NLVL] = {
    16.f, 20.f, 25.f,  32.f,  40.f,  50.f,  64.f,  80.f,
    101.f, 128.f, 161.f, 203.f, 256.f, 322.f, 406.f, 512.f};

// Packed 2-wide lerp in the reference's op order: a*(1-w) + b*w.
// Both features share the weight -> v_pk_fma_f32 / v_pk_mul_f32 candidates.
__device__ __forceinline__ v2f lerp2(v2f a, v2f b, float w) {
  return a * (1.0f - w) + b * w;
}

// UNIT=true: box_min==0, box_max==1 (wave-uniform predicate checked by caller).
// All box terms become literals: grid folds to fl(1/res) at compile time; the
// six power-of-two levels' floor-divides legally become exact multiplies.
template <bool UNIT>
__device__ __forceinline__ void encode_levels(
    float px, float py, float pz,
    float b0x_, float b0y_, float b0z_,
    float b1x_, float b1y_, float b1z_,
    const float* __restrict__ emb, const float* __restrict__ lw,
    v2f acc[NLVL]) {
  const float b0x = UNIT ? 0.0f : b0x_;
  const float b0y = UNIT ? 0.0f : b0y_;
  const float b0z = UNIT ? 0.0f : b0z_;
  const float b1x = UNIT ? 1.0f : b1x_;
  const float b1y = UNIT ? 1.0f : b1y_;
  const float b1z = UNIT ? 1.0f : b1z_;

  // xc = clip(x, bmin, bmax)
  float xcx = fminf(fmaxf(px, b0x), b1x);
  float xcy = fminf(fmaxf(py, b0y), b1y);
  float xcz = fminf(fmaxf(pz, b0z), b1z);

  // Loop-invariant numerators of the floor quotients.
  float dxx = xcx - b0x;
  float dxy = xcy - b0y;
  float dxz = xcz - b0z;

#pragma unroll
  for (int i = 0; i < NLVL; ++i) {
    const float res = kRes[i];
    // grid = (bmax - bmin)/res. UNIT path: constant-folded to fl(1/res).
    float gx = (b1x - b0x) / res;
    float gy = (b1y - b0y) / res;
    float gz = (b1z - b0z) / res;

    // bl = floor((xc - bmin)/grid): the ONLY discontinuity -> exact IEEE ops.
    // (UNIT + pow2 res: fdiv by exact-pow2 constant becomes an exact multiply.)
    float fx = floorf(dxx / gx);
    float fy = floorf(dxy / gy);
    float fz = floorf(dxz / gz);

    // vmin = bl*grid + bmin ; w = (x - vmin)/((vmin+grid) - vmin).
    // w is smooth -> 1-ulp v_rcp_f32 + v_mul instead of a ~10-op IEEE divide.
    float vmx = fx * gx + b0x;
    float vmy = fy * gy + b0y;
    float vmz = fz * gz + b0z;
    float wx = (px - vmx) * __builtin_amdgcn_rcpf((vmx + gx) - vmx);
    float wy = (py - vmy) * __builtin_amdgcn_rcpf((vmy + gy) - vmy);
    float wz = (pz - vmz) * __builtin_amdgcn_rcpf((vmz + gz) - vmz);

    uint32_t ix = (uint32_t)(int)fx;
    uint32_t iy = (uint32_t)(int)fy;
    uint32_t iz = (uint32_t)(int)fz;

    // Spatial hash; (c+1)*P == c*P + P in uint32 arithmetic.
    uint32_t hx0 = ix;                  uint32_t hx1 = hx0 + 1u;
    uint32_t hy0 = iy * 2654435761u;    uint32_t hy1 = hy0 + 2654435761u;
    uint32_t hz0 = iz * 805459861u;     uint32_t hz1 = hz0 + 805459861u;

    const v2f* __restrict__ tbl = (const v2f*)emb + ((size_t)i << LOG2T);

    // 8 independent b64 gathers -> one clause + one s_wait_loadcnt; default RT
    // hint keeps the 64 MB table resident in the 192 MB L2; saddr+scale_offset
    // addressing keeps per-gather VALU at the hash ops only.
    v2f e000 = tbl[(hx0 ^ hy0 ^ hz0) & TMASK];
    v2f e001 = tbl[(hx0 ^ hy0 ^ hz1) & TMASK];
    v2f e010 = tbl[(hx0 ^ hy1 ^ hz0) & TMASK];
    v2f e011 = tbl[(hx0 ^ hy1 ^ hz1) & TMASK];
    v2f e100 = tbl[(hx1 ^ hy0 ^ hz0) & TMASK];
    v2f e101 = tbl[(hx1 ^ hy0 ^ hz1) & TMASK];
    v2f e110 = tbl[(hx1 ^ hy1 ^ hz0) & TMASK];
    v2f e111 = tbl[(hx1 ^ hy1 ^ hz1) & TMASK];

    // Trilinear blend on packed feature pairs (x, then y, then z).
    v2f c00 = lerp2(e000, e100, wx);
    v2f c01 = lerp2(e001, e101, wx);
    v2f c10 = lerp2(e010, e110, wx);
    v2f c11 = lerp2(e011, e111, wx);
    v2f c0  = lerp2(c00, c10, wy);
    v2f c1  = lerp2(c01, c11, wy);
    v2f c   = lerp2(c0, c1, wz);

    // Smooth gate: fast exp + rcp (sigmoid(0)=0.5 is still exact here).
    float gate = __builtin_amdgcn_rcpf(1.0f + __expf(-lw[i]));
    acc[i] = c * gate;
  }
}

__global__ __launch_bounds__(256) void hashgrid_encode_kernel(
    const float* __restrict__ x,     // [N,3]
    const float* __restrict__ emb,   // [16, 2^19, 2]
    const float* __restrict__ lw,    // [16]
    const float* __restrict__ bmin,  // [3]
    const float* __restrict__ bmax,  // [3]
    float* __restrict__ out,         // [N,32]
    int n) {
  int p = blockIdx.x * blockDim.x + threadIdx.x;
  if (p >= n) return;

  // x is read exactly once -> NT loads (TH_LOAD_NT), don't pollute L2.
  const float* xp = x + 3 * (size_t)p;
  float px = __builtin_nontemporal_load(xp + 0);
  float py = __builtin_nontemporal_load(xp + 1);
  float pz = __builtin_nontemporal_load(xp + 2);

  float b0x = bmin[0], b0y = bmin[1], b0z = bmin[2];
  float b1x = bmax[0], b1y = bmax[1], b1z = bmax[2];

  v2f acc[NLVL];

  // Wave-uniform branch (scalar condition): specialized unit-box path folds the
  // grid constants at compile time; generic path preserves full correctness.
  bool unit = (b0x == 0.0f) & (b0y == 0.0f) & (b0z == 0.0f) &
              (b1x == 1.0f) & (b1y == 1.0f) & (b1z == 1.0f);
  if (unit) {
    encode_levels<true>(px, py, pz, b0x, b0y, b0z, b1x, b1y, b1z, emb, lw, acc);
  } else {
    encode_levels<false>(px, py, pz, b0x, b0y, b0z, b1x, b1y, b1z, emb, lw, acc);
  }

  // Each lane owns exactly one 128B output line; stream it with NT b128 stores
  // so the 128 MB output never evicts the embedding tables from L2.
  float* op = out + (size_t)p * (2 * NLVL);
#pragma unroll
  for (int q = 0; q < NLVL / 2; ++q) {
    v4f v = {acc[2 * q].x, acc[2 * q].y, acc[2 * q + 1].x, acc[2 * q + 1].y};
    __builtin_nontemporal_store(v, (v4f*)(op + 4 * q));
  }
}

extern "C" void kernel_launch(void* const* d_in, const int* in_sizes, int n_in,
                              void* d_out, int out_size, void* d_ws,
                              size_t ws_size, hipStream_t stream) {
  const float* x    = (const float*)d_in[0];  // [N,3]
  const float* emb  = (const float*)d_in[1];  // [16, 2^19, 2]
  const float* lw   = (const float*)d_in[2];  // [16]
  const float* bmin = (const float*)d_in[3];  // [3]
  const float* bmax = (const float*)d_in[4];  // [3]
  float* out = (float*)d_out;

  int n = in_sizes[0] / 3;
  dim3 block(256);                 // 8 wave32 waves per workgroup
  dim3 grid((n + 255) / 256);
  hashgrid_encode_kernel<<<grid, block, 0, stream>>>(x, emb, lw, bmin, bmax,
                                                     out, n);
}